// ResNet50Conv5ROIFeatureExtractor_37769942401335
// MI455X (gfx1250) — compile-verified
//
#include <hip/hip_runtime.h>
#include <hip/hip_bf16.h>

typedef float v2f __attribute__((ext_vector_type(2)));
typedef float v8f __attribute__((ext_vector_type(8)));

#define HF 50
#define WF 76
#define CF 1024
#define HWF (HF * WF)          // 3800
#define NROI 128
#define MTOT (NROI * 49)       // 6272
#define OC 2048

#define FLAG_RELU 1
#define FLAG_NCHW 2

static __device__ __forceinline__ v8f wmma4(v2f a, v2f b, v8f c) {
    // D = A(16x4 f32) x B(4x16 f32) + C(16x16 f32)
    return __builtin_amdgcn_wmma_f32_16x16x4_f32(false, a, false, b, (short)0, c, false, false);
}

// CDNA5 async global -> LDS (ASYNCcnt-tracked, no VGPR round trip).
// lds_off: byte offset of destination in LDS; base: uniform 64-bit base (SGPR pair);
// voff: per-lane signed 32-bit byte offset.
static __device__ __forceinline__ void async_ld128(unsigned lds_off, const float* base, int voff) {
    asm volatile("global_load_async_to_lds_b128 %0, %1, %2"
                 :: "v"(lds_off), "v"(voff), "s"(base)
                 : "memory");
}
static __device__ __forceinline__ void wait_async0() {
    asm volatile("s_wait_asynccnt 0x0" ::: "memory");
}

// ---------------- features NCHW -> NHWC transpose ----------------
__global__ __launch_bounds__(256)
void feat_transpose_kernel(const float* __restrict__ src, float* __restrict__ dst) {
    __shared__ float tile[32][33];
    const int b  = blockIdx.z;
    const int c0 = blockIdx.y * 32;
    const int p0 = blockIdx.x * 32;
    const int tx = threadIdx.x;   // 0..31
    const int ty = threadIdx.y;   // 0..7
#pragma unroll
    for (int i = 0; i < 32; i += 8) {
        int c = c0 + ty + i;
        int p = p0 + tx;
        float v = 0.f;
        if (p < HWF) v = src[((size_t)b * CF + c) * HWF + p];
        tile[ty + i][tx] = v;
    }
    __syncthreads();
#pragma unroll
    for (int i = 0; i < 32; i += 8) {
        int p = p0 + ty + i;
        int c = c0 + tx;
        if (p < HWF) dst[((size_t)b * HWF + p) * CF + c] = tile[tx][ty + i];
    }
}

// ---------------- ROI align at stride-2 bins -> [6272][1024] NHWC ----------------
__global__ __launch_bounds__(256)
void roi_align_kernel(const float* __restrict__ featT,
                      const float* __restrict__ boxes,
                      const int* __restrict__ roib,
                      float* __restrict__ out) {
    const int m  = blockIdx.x;          // roi*49 + oy*7 + ox
    const int n  = m / 49;
    const int p  = m - n * 49;
    const int oy = p / 7, ox = p - (p / 7) * 7;

    const float sc = 1.0f / 16.0f;
    const float x1 = boxes[n * 4 + 0] * sc;
    const float y1 = boxes[n * 4 + 1] * sc;
    const float x2 = boxes[n * 4 + 2] * sc;
    const float y2 = boxes[n * 4 + 3] * sc;
    const int   b  = roib[n];
    const float bin_h = fmaxf(y2 - y1, 1.0f) * (1.0f / 14.0f);
    const float bin_w = fmaxf(x2 - x1, 1.0f) * (1.0f / 14.0f);

    int yl[2], yh[2], xl[2], xh[2];
    float w0y[2], w1y[2], w0x[2], w1x[2];
    bool vy[2], vx[2];
#pragma unroll
    for (int s = 0; s < 2; ++s) {
        float g = ((float)(4 * oy + s) + 0.5f) * 0.5f;   // stride-2 bin: t = 4*oy + s
        float y = y1 + g * bin_h;
        vy[s] = (y >= -1.0f) && (y <= (float)HF);
        float yc = fminf(fmaxf(y, 0.0f), (float)(HF - 1));
        float fl = floorf(yc);
        yl[s] = (int)fl;
        yh[s] = min(yl[s] + 1, HF - 1);
        w1y[s] = yc - fl;
        w0y[s] = 1.0f - w1y[s];
        g = ((float)(4 * ox + s) + 0.5f) * 0.5f;
        float x = x1 + g * bin_w;
        vx[s] = (x >= -1.0f) && (x <= (float)WF);
        float xc = fminf(fmaxf(x, 0.0f), (float)(WF - 1));
        fl = floorf(xc);
        xl[s] = (int)fl;
        xh[s] = min(xl[s] + 1, WF - 1);
        w1x[s] = xc - fl;
        w0x[s] = 1.0f - w1x[s];
    }

    const float* fb = featT + (size_t)b * HWF * CF;
    for (int c = threadIdx.x; c < CF; c += 256) {
        float acc = 0.f;
#pragma unroll
        for (int sy = 0; sy < 2; ++sy) {
#pragma unroll
            for (int sx = 0; sx < 2; ++sx) {
                if (vy[sy] && vx[sx]) {
                    const float* r0 = fb + (size_t)(yl[sy] * WF) * CF;
                    const float* r1 = fb + (size_t)(yh[sy] * WF) * CF;
                    float f00 = r0[(size_t)xl[sx] * CF + c];
                    float f01 = r0[(size_t)xh[sx] * CF + c];
                    float f10 = r1[(size_t)xl[sx] * CF + c];
                    float f11 = r1[(size_t)xh[sx] * CF + c];
                    acc += (f00 * w0x[sx] + f01 * w1x[sx]) * w0y[sy] +
                           (f10 * w0x[sx] + f11 * w1x[sx]) * w1y[sy];
                }
            }
        }
        out[(size_t)m * CF + c] = acc * 0.25f;
    }
}

// ---------------- w2 repack: [n][c*9+q] -> [n][q*512+c] ----------------
__global__ __launch_bounds__(256)
void w2_pack_kernel(const float* __restrict__ src, float* __restrict__ dst) {
    const int idx = blockIdx.x * 256 + threadIdx.x;   // over 512*4608
    const int n = idx / 4608;
    const int k = idx - n * 4608;
    const int q = k >> 9;
    const int c = k & 511;
    dst[idx] = src[(size_t)n * 4608 + c * 9 + q];
}

// ---------------- WMMA GEMM: out[M,N] = epilogue(A[M,K] * W[N,K]^T) ----------------
// Block tile 128x128, 8 wave32s (4 along M x 2 along N), wave tile 32x64 (2x4 WMMA tiles).
// Double-buffered LDS, async global->LDS staging.
// AMODE 0: A[m,k] = A[m*K + k]
// AMODE 1: im2col 3x3 pad 1 from A=[6272][512] NHWC; k = q*512 + c (W pre-packed to match)
template <int AMODE>
__global__ __launch_bounds__(256)
void gemm_kernel(const float* __restrict__ A, const float* __restrict__ W,
                 const float* __restrict__ scale, const float* __restrict__ bias,
                 const float* __restrict__ resid, float* __restrict__ out,
                 int K, int N, int flags) {
    __shared__ float shA[2][128][20];   // [buf][m][k] pad 20: conflict-free 8B b64 frags
    __shared__ float shB[2][128][20];   // [buf][n][k]

    const int tid   = threadIdx.x;
    const int lane  = tid & 31;
    const int wid   = tid >> 5;
    const int waveM = wid & 3;      // 4 waves along M (32 rows each)
    const int waveN = wid >> 2;     // 2 waves along N (64 cols each)
    const int r     = lane & 15;
    const int kq    = lane >> 4;
    const int mBlk  = blockIdx.y * 128;
    const int nBlk  = blockIdx.x * 128;

    // Loader: each thread stages two 16B chunks per tile: rows lrow, lrow+64, cols lcol..lcol+3
    const int lrow = tid >> 2;          // 0..63
    const int lcol = (tid & 3) * 4;     // 0,4,8,12

    int aRowOff[2], bRowOff[2];
    int roiI[2], oyr[2], oxr[2];
#pragma unroll
    for (int h = 0; h < 2; ++h) {
        const int row = h * 64 + lrow;
        if (AMODE == 0) {
            aRowOff[h] = (mBlk + row) * K + lcol;
        } else {
            const int mg = mBlk + row;
            roiI[h] = mg / 49;
            const int pp = mg - roiI[h] * 49;
            oyr[h] = pp / 7;
            oxr[h] = pp - oyr[h] * 7;
        }
        bRowOff[h] = (nBlk + row) * K + lcol;
    }
    unsigned ldsA[2][2], ldsB[2][2];
#pragma unroll
    for (int bf = 0; bf < 2; ++bf)
#pragma unroll
        for (int h = 0; h < 2; ++h) {
            ldsA[bf][h] = (unsigned)(size_t)&shA[bf][h * 64 + lrow][lcol];
            ldsB[bf][h] = (unsigned)(size_t)&shB[bf][h * 64 + lrow][lcol];
        }

    v8f acc[2][4];
#pragma unroll
    for (int i = 0; i < 2; ++i)
#pragma unroll
        for (int j = 0; j < 4; ++j) {
            v8f z = {0.f, 0.f, 0.f, 0.f, 0.f, 0.f, 0.f, 0.f};
            acc[i][j] = z;
        }

    const int nStages = K >> 4;

    auto issue = [&](int st) {
        const int k0  = st << 4;
        const int buf = st & 1;
#pragma unroll
        for (int h = 0; h < 2; ++h)
            async_ld128(ldsB[buf][h], W, (bRowOff[h] + k0) * 4);
        if (AMODE == 0) {
#pragma unroll
            for (int h = 0; h < 2; ++h)
                async_ld128(ldsA[buf][h], A, (aRowOff[h] + k0) * 4);
        } else {
            const int q  = k0 >> 9;               // tap 0..8 (stage-uniform)
            const int dy = q / 3, dx = q - (q / 3) * 3;
            const int c0 = (k0 & 511) + lcol;
#pragma unroll
            for (int h = 0; h < 2; ++h) {
                const int iy = oyr[h] + dy - 1;
                const int ix = oxr[h] + dx - 1;
                if ((unsigned)iy < 7u && (unsigned)ix < 7u) {
                    async_ld128(ldsA[buf][h], A,
                                ((roiI[h] * 49 + iy * 7 + ix) * 512 + c0) * 4);
                } else {
                    float4 z = make_float4(0.f, 0.f, 0.f, 0.f);
                    *(float4*)&shA[buf][h * 64 + lrow][lcol] = z;   // zero-pad halo
                }
            }
        }
    };

    issue(0);
    for (int st = 0; st < nStages; ++st) {
        wait_async0();          // my loads for stage st complete
        __syncthreads();        // all waves loaded st; all waves done computing st-1
        if (st + 1 < nStages) issue(st + 1);   // overlaps with compute below
        const int cur = st & 1;
#pragma unroll
        for (int ks = 0; ks < 4; ++ks) {
            const int kb = ks * 4 + kq * 2;     // lane's K pair {kb, kb+1}
            v2f a0 = *(const v2f*)&shA[cur][waveM * 32 + r][kb];
            v2f a1 = *(const v2f*)&shA[cur][waveM * 32 + 16 + r][kb];
            v2f b0 = *(const v2f*)&shB[cur][waveN * 64 + r][kb];
            v2f b1 = *(const v2f*)&shB[cur][waveN * 64 + 16 + r][kb];
            v2f b2 = *(const v2f*)&shB[cur][waveN * 64 + 32 + r][kb];
            v2f b3 = *(const v2f*)&shB[cur][waveN * 64 + 48 + r][kb];
            acc[0][0] = wmma4(a0, b0, acc[0][0]);
            acc[0][1] = wmma4(a0, b1, acc[0][1]);
            acc[0][2] = wmma4(a0, b2, acc[0][2]);
            acc[0][3] = wmma4(a0, b3, acc[0][3]);
            acc[1][0] = wmma4(a1, b0, acc[1][0]);
            acc[1][1] = wmma4(a1, b1, acc[1][1]);
            acc[1][2] = wmma4(a1, b2, acc[1][2]);
            acc[1][3] = wmma4(a1, b3, acc[1][3]);
        }
    }

    // ---- epilogue: scale/bias (+residual) (+relu), NHWC or NCHW store ----
    const bool relu = (flags & FLAG_RELU) != 0;
    const bool nchw = (flags & FLAG_NCHW) != 0;
#pragma unroll
    for (int tm = 0; tm < 2; ++tm) {
#pragma unroll
        for (int tn = 0; tn < 4; ++tn) {
            const int nCol  = nBlk + waveN * 64 + tn * 16 + r;
            const int mRow0 = mBlk + waveM * 32 + tm * 16 + kq * 8;
            const float s  = scale[nCol];
            const float tb = bias[nCol];
#pragma unroll
            for (int v = 0; v < 8; ++v) {
                const int mr = mRow0 + v;
                float val = acc[tm][tn][v] * s + tb;
                if (resid) val += resid[(size_t)mr * N + nCol];
                if (relu) val = fmaxf(val, 0.f);
                if (nchw) {
                    const int roiO = mr / 49;
                    const int pp   = mr - roiO * 49;
                    out[((size_t)roiO * N + nCol) * 49 + pp] = val;
                } else {
                    out[(size_t)mr * N + nCol] = val;
                }
            }
        }
    }
}

extern "C" void kernel_launch(void* const* d_in, const int* in_sizes, int n_in,
                              void* d_out, int out_size, void* d_ws, size_t ws_size,
                              hipStream_t stream) {
    const float* feats = (const float*)d_in[0];
    const float* boxes = (const float*)d_in[1];
    const int*   roib  = (const int*)d_in[2];
    // params leaves in sorted pytree order: block0 {s1,s2,s3,sd,t1,t2,t3,td,w1,w2,w3,wd},
    // block1/2 {s1,s2,s3,t1,t2,t3,w1,w2,w3}
    const float* s1_0 = (const float*)d_in[3],  *s2_0 = (const float*)d_in[4];
    const float* s3_0 = (const float*)d_in[5],  *sd_0 = (const float*)d_in[6];
    const float* t1_0 = (const float*)d_in[7],  *t2_0 = (const float*)d_in[8];
    const float* t3_0 = (const float*)d_in[9],  *td_0 = (const float*)d_in[10];
    const float* w1_0 = (const float*)d_in[11], *w2_0 = (const float*)d_in[12];
    const float* w3_0 = (const float*)d_in[13], *wd_0 = (const float*)d_in[14];
    const float* s1_1 = (const float*)d_in[15], *s2_1 = (const float*)d_in[16];
    const float* s3_1 = (const float*)d_in[17];
    const float* t1_1 = (const float*)d_in[18], *t2_1 = (const float*)d_in[19];
    const float* t3_1 = (const float*)d_in[20];
    const float* w1_1 = (const float*)d_in[21], *w2_1 = (const float*)d_in[22];
    const float* w3_1 = (const float*)d_in[23];
    const float* s1_2 = (const float*)d_in[24], *s2_2 = (const float*)d_in[25];
    const float* s3_2 = (const float*)d_in[26];
    const float* t1_2 = (const float*)d_in[27], *t2_2 = (const float*)d_in[28];
    const float* t3_2 = (const float*)d_in[29];
    const float* w1_2 = (const float*)d_in[30], *w2_2 = (const float*)d_in[31];
    const float* w3_2 = (const float*)d_in[32];
    (void)in_sizes; (void)n_in; (void)out_size;

    const size_t SZ_FEATT = (size_t)2 * HWF * CF;    // 7,782,400
    const size_t SZ_ROI   = (size_t)MTOT * CF;       // 6,422,528
    const size_t SZ_PQ    = (size_t)MTOT * OC;       // 12,845,056
    const size_t SZ_AB    = (size_t)MTOT * 512;      // 3,211,264
    const size_t SZ_W2    = (size_t)512 * 4608;      // 2,359,296
    const size_t needB = (SZ_FEATT + SZ_ROI + 2 * SZ_PQ + 2 * SZ_AB) * sizeof(float);
    if (ws_size < needB) return;

    float* featT = (float*)d_ws;
    float* roi   = featT + SZ_FEATT;
    float* P     = roi + SZ_ROI;
    float* Q     = P + SZ_PQ;
    float* ab    = Q + SZ_PQ;
    float* bb    = ab + SZ_AB;
    // featT is dead after roi_align; reuse it for the packed 3x3 weights (3*2.36M < 7.78M).
    // Stream ordering makes this aliasing safe.
    float* w2r_0 = featT;
    float* w2r_1 = featT + SZ_W2;
    float* w2r_2 = featT + 2 * SZ_W2;
    float* outF  = (float*)d_out;

    dim3 thr256(256);
    const dim3 gN512(512 / 128, MTOT / 128);
    const dim3 gN2048(OC / 128, MTOT / 128);

    // 1) NCHW -> NHWC features (channel-coalesced ROI gathers)
    feat_transpose_kernel<<<dim3((HWF + 31) / 32, CF / 32, 2), dim3(32, 8), 0, stream>>>(feats, featT);
    // 2) ROI align at the 49 stride-2 bins only -> roi [6272][1024]
    roi_align_kernel<<<dim3(MTOT), thr256, 0, stream>>>(featT, boxes, roib, roi);
    // 3) repack 3x3 weights to k = q*512+c order (featT region now dead)
    w2_pack_kernel<<<dim3((512 * 4608) / 256), thr256, 0, stream>>>(w2_0, w2r_0);
    w2_pack_kernel<<<dim3((512 * 4608) / 256), thr256, 0, stream>>>(w2_1, w2r_1);
    w2_pack_kernel<<<dim3((512 * 4608) / 256), thr256, 0, stream>>>(w2_2, w2r_2);

    // ---- block0 ----
    gemm_kernel<0><<<gN2048, thr256, 0, stream>>>(roi, wd_0, sd_0, td_0, nullptr, Q, 1024, OC, 0);
    gemm_kernel<0><<<gN512, thr256, 0, stream>>>(roi, w1_0, s1_0, t1_0, nullptr, ab, 1024, 512, FLAG_RELU);
    gemm_kernel<1><<<gN512, thr256, 0, stream>>>(ab, w2r_0, s2_0, t2_0, nullptr, bb, 4608, 512, FLAG_RELU);
    gemm_kernel<0><<<gN2048, thr256, 0, stream>>>(bb, w3_0, s3_0, t3_0, Q, P, 512, OC, FLAG_RELU);
    // ---- block1 (identity = P) ----
    gemm_kernel<0><<<gN512, thr256, 0, stream>>>(P, w1_1, s1_1, t1_1, nullptr, ab, 2048, 512, FLAG_RELU);
    gemm_kernel<1><<<gN512, thr256, 0, stream>>>(ab, w2r_1, s2_1, t2_1, nullptr, bb, 4608, 512, FLAG_RELU);
    gemm_kernel<0><<<gN2048, thr256, 0, stream>>>(bb, w3_1, s3_1, t3_1, P, Q, 512, OC, FLAG_RELU);
    // ---- block2 (identity = Q), final GEMM writes d_out NCHW ----
    gemm_kernel<0><<<gN512, thr256, 0, stream>>>(Q, w1_2, s1_2, t1_2, nullptr, ab, 2048, 512, FLAG_RELU);
    gemm_kernel<1><<<gN512, thr256, 0, stream>>>(ab, w2r_2, s2_2, t2_2, nullptr, bb, 4608, 512, FLAG_RELU);
    gemm_kernel<0><<<gN2048, thr256, 0, stream>>>(bb, w3_2, s3_2, t3_2, Q, outF, 512, OC, FLAG_RELU | FLAG_NCHW);
}